// GlottalComplexConjLPCEncoder_46523085750317
// MI455X (gfx1250) — compile-verified
//
#include <hip/hip_runtime.h>
#include <math.h>

typedef __attribute__((ext_vector_type(2))) float v2f;
typedef __attribute__((ext_vector_type(8))) float v8f;
typedef __attribute__((ext_vector_type(4))) int   v4i;

#define M_TOTAL 32000      // B*F = 32*1000
#define K_DIM   1024
#define N_OUT   175        // 1+22+1+22+1+128
#define NT      11         // N tiles of 16 (176 padded)
#define KC      32         // K chunk staged in LDS
#define HS      36         // LDS row stride (32 + 4 pad) in floats
#define ROWS_PER_BLOCK 128 // 8 waves * 16 rows

// output layout (flat, in reference return order)
#define OFF_F0 0u
#define OFF_HF 32000u            // h_feat: 32000*128
#define OFF_VG 4128000u          // voice_gain
#define OFF_VL 4160000u          // voice_lpc: 32000*22
#define OFF_NG 4864000u          // noise_gain
#define OFF_NL 4896000u          // noise_lpc

#define LOG_F0_MIN 4.3820266346738812f
#define LOG_F0_MAX 6.9077552789821368f

#if __has_builtin(__builtin_amdgcn_global_load_async_to_lds_b128) && \
    __has_builtin(__builtin_amdgcn_s_wait_asynccnt)
#define USE_ASYNC_LDS 1
typedef __attribute__((address_space(1))) v4i as1_v4i;   // global
typedef __attribute__((address_space(3))) v4i as3_v4i;   // LDS
__device__ __forceinline__ void async_copy_b128(const float* gsrc, float* lds_dst)
{
    __builtin_amdgcn_global_load_async_to_lds_b128(
        (as1_v4i*)gsrc, (as3_v4i*)lds_dst, /*offset=*/0, /*cpol=*/0);
}
#else
#define USE_ASYNC_LDS 0
#endif

__global__ __launch_bounds__(256)
void gemm_wmma_f32(const float* __restrict__ h, const float* __restrict__ W,
                   const float* __restrict__ bias, float* __restrict__ out,
                   float* __restrict__ ws)
{
    __shared__ float sh_h[ROWS_PER_BLOCK * HS];   // 18.4 KB
    __shared__ float sh_w[176 * HS];              // 25.3 KB

    const int tid  = threadIdx.x;
    const int wave = tid >> 5;
    const int lane = tid & 31;
    const int half = lane >> 4;   // 0 or 1
    const int lrow = lane & 15;
    const int m0   = blockIdx.x * ROWS_PER_BLOCK;

    // accumulators: 11 tiles of 16x16, bias folded in (per-column bias)
    v8f acc[NT];
#pragma unroll
    for (int nt = 0; nt < NT; ++nt) {
        const int n = nt * 16 + lrow;
        const float bv = (n < N_OUT) ? bias[n] : 0.0f;
        acc[nt] = (v8f){bv, bv, bv, bv, bv, bv, bv, bv};
    }

    // zero the padded W row (n == 175) once; loader never touches it
    if (tid < HS) sh_w[175 * HS + tid] = 0.0f;

    const float* aBase = &sh_h[(wave * 16 + lrow) * HS + 2 * half];
    const float* bBase = &sh_w[lrow * HS + 2 * half];

    for (int kc = 0; kc < K_DIM; kc += KC) {
        __syncthreads();
        // stage h: 128 rows x 32 K = 1024 x b128, 4 per thread, coalesced
#pragma unroll
        for (int i = 0; i < 4; ++i) {
            const int idx = tid + i * 256;
            const int r   = idx >> 3;
            const int ko  = (idx & 7) << 2;
            const float* src = &h[(size_t)(m0 + r) * K_DIM + kc + ko];
            float*       dst = &sh_h[r * HS + ko];
#if USE_ASYNC_LDS
            async_copy_b128(src, dst);
#else
            *(float4*)dst = *(const float4*)src;
#endif
        }
        // stage W: 175 rows x 32 K = 1400 x b128
#pragma unroll
        for (int i = 0; i < 6; ++i) {
            const int idx = tid + i * 256;
            if (idx < 1400) {
                const int r  = idx >> 3;
                const int ko = (idx & 7) << 2;
                const float* src = &W[(size_t)r * K_DIM + kc + ko];
                float*       dst = &sh_w[r * HS + ko];
#if USE_ASYNC_LDS
                async_copy_b128(src, dst);
#else
                *(float4*)dst = *(const float4*)src;
#endif
            }
        }
#if USE_ASYNC_LDS
        __builtin_amdgcn_s_wait_asynccnt(0);
#endif
        __syncthreads();

        // 8 K-steps of 4 per chunk; 11 WMMAs per step
#pragma unroll
        for (int ks = 0; ks < KC; ks += 4) {
            const v2f a = *(const v2f*)(aBase + ks);
#pragma unroll
            for (int nt = 0; nt < NT; ++nt) {
                const v2f b = *(const v2f*)(bBase + nt * 16 * HS + ks);
                acc[nt] = __builtin_amdgcn_wmma_f32_16x16x4_f32(
                    false, a, false, b, (short)0, acc[nt], false, false);
            }
        }
    }

    // epilogue: C/D layout -> VGPR j is row (j + 8*half), lane lrow is column
#pragma unroll
    for (int nt = 0; nt < NT; ++nt) {
        const int n = nt * 16 + lrow;
        if (n >= N_OUT) continue;
#pragma unroll
        for (int j = 0; j < 8; ++j) {
            const int m = m0 + wave * 16 + j + 8 * half;
            const float v = acc[nt][j];
            if (n >= 47)
                out[OFF_HF + (size_t)m * 128u + (unsigned)(n - 47)] = v;   // h_feat direct
            else
                ws[(size_t)m * 47u + (unsigned)n] = v;                    // front cols to scratch
        }
    }
}

__device__ __forceinline__ float sigmoidf_(float x) { return 1.0f / (1.0f + expf(-x)); }

__device__ void lpc_from_logits(const float* __restrict__ lg, float* __restrict__ dst)
{
    float p[23];
    p[0] = 1.0f;
#pragma unroll
    for (int i = 1; i < 23; ++i) p[i] = 0.0f;
#pragma unroll
    for (int s = 0; s < 11; ++s) {
        const float mag = 0.99f * sigmoidf_(lg[2 * s]);
        const float ph  = 3.14159265358979323846f * sigmoidf_(lg[2 * s + 1]);
        const float a1  = -2.0f * mag * cosf(ph);
        const float a2  = mag * mag;
        // poly *= (1 + a1 z + a2 z^2), truncated to length 23 (backward in-place)
#pragma unroll
        for (int j = 22; j >= 2; --j) p[j] += a1 * p[j - 1] + a2 * p[j - 2];
        p[1] += a1 * p[0];
    }
#pragma unroll
    for (int i = 0; i < 22; ++i) dst[i] = p[i + 1];
}

__global__ __launch_bounds__(256)
void postprocess(const float* __restrict__ ws, float* __restrict__ out)
{
    const int m = blockIdx.x * blockDim.x + threadIdx.x;
    if (m >= M_TOTAL) return;
    const float* r = ws + (size_t)m * 47u;

    out[OFF_F0 + m] = expf(sigmoidf_(r[0]) * (LOG_F0_MAX - LOG_F0_MIN) + LOG_F0_MIN);
    out[OFF_VG + m] = expf(r[23]);
    out[OFF_NG + m] = expf(r[46]);

    float tmp[22];
    lpc_from_logits(r + 1, tmp);
#pragma unroll
    for (int i = 0; i < 22; ++i) out[OFF_VL + (size_t)m * 22u + i] = tmp[i];
    lpc_from_logits(r + 24, tmp);
#pragma unroll
    for (int i = 0; i < 22; ++i) out[OFF_NL + (size_t)m * 22u + i] = tmp[i];
}

extern "C" void kernel_launch(void* const* d_in, const int* in_sizes, int n_in,
                              void* d_out, int out_size, void* d_ws, size_t ws_size,
                              hipStream_t stream)
{
    const float* h = (const float*)d_in[0];   // (32,1000,1024)
    const float* W = (const float*)d_in[1];   // (175,1024)
    const float* b = (const float*)d_in[2];   // (175,)
    float* out = (float*)d_out;
    float* ws  = (float*)d_ws;                // needs 32000*47*4 = ~6.0 MB

    gemm_wmma_f32<<<M_TOTAL / ROWS_PER_BLOCK, 256, 0, stream>>>(h, W, b, out, ws);
    postprocess<<<(M_TOTAL + 255) / 256, 256, 0, stream>>>(ws, out);
}